// Wn0110C2_38268158608163
// MI455X (gfx1250) — compile-verified
//
#include <hip/hip_runtime.h>
#include <math.h>

// ---------------- problem constants ----------------
#define NHh   16
#define NSs   1024
#define NTt   730
static constexpr int       NSNH = NSs * NHh;                 // 16384
static constexpr long long MTOT = (long long)NTt * NSs;      // 747520
static constexpr long long MH   = MTOT * NHh;                // 11,960,320

// ---------------- workspace layout (float offsets) ----------------
static constexpr long long OFF_PL  = 0;
static constexpr long long OFF_EV  = 1 * MH;
static constexpr long long OFF_VM  = 2 * MH;
static constexpr long long OFF_RS0 = 3 * MH;
static constexpr long long OFF_RS1 = 4 * MH;
static constexpr long long OFF_RG0 = 5 * MH;
static constexpr long long OFF_RG1 = 6 * MH;
static constexpr long long OFF_PS  = 7 * MH;                  // [MTOT]
static constexpr long long OFF_PAR = 7 * MH + MTOT;           // 13 arrays x NSNH
static constexpr long long OFF_RR  = OFF_PAR + 13LL * NSNH;   // [NSNH*8]

typedef __attribute__((ext_vector_type(2))) float v2f;
typedef __attribute__((ext_vector_type(8))) float v8f;

static __device__ inline v8f splat8(float v) {
  v8f r = {v, v, v, v, v, v, v, v};
  return r;
}
static __device__ inline v8f wmma4(v2f a, v2f b, v8f c) {
  // D = A(16x4,f32) x B(4x16,f32) + C(16x16,f32)
  return __builtin_amdgcn_wmma_f32_16x16x4_f32(false, a, false, b, (short)0, c,
                                               false, false);
}
static __device__ inline v2f ldv2(const float* p) {
  // p is guaranteed 8-byte aligned by construction (even dword offsets)
  return *(const v2f*)p;
}
static __device__ inline float sigmoidf_(float x) { return 1.f / (1.f + expf(-x)); }

// =====================================================================
// Kernel 1: per-site MLPs (fcW, fcC, fcR) + derived static parameters
// =====================================================================
__global__ __launch_bounds__(256) void k_site(
    const float* __restrict__ xc,
    const float* __restrict__ Rw1, const float* __restrict__ Rb1,
    const float* __restrict__ Rw2, const float* __restrict__ Rb2,
    const float* __restrict__ Ww1, const float* __restrict__ Wb1,
    const float* __restrict__ Ww2, const float* __restrict__ Wb2,
    const float* __restrict__ Cw1, const float* __restrict__ Cb1,
    const float* __restrict__ Cw2, const float* __restrict__ Cb2,
    float* __restrict__ ws) {
  __shared__ float xcs[32];
  __shared__ float hid[256];
  __shared__ float ow[112];   // fcW raw outputs
  __shared__ float oc[64];    // fcC raw outputs
  __shared__ float orr[128];  // fcR raw outputs
  const int tid = threadIdx.x;
  const int s = blockIdx.x;
  if (tid < 32) xcs[tid] = xc[s * 32 + tid];
  __syncthreads();

  // ---- fcW ----
  {
    float a = Wb1[tid];
#pragma unroll
    for (int k = 0; k < 32; k++) a = fmaf(xcs[k], Ww1[k * 256 + tid], a);
    hid[tid] = tanhf(a);
  }
  __syncthreads();
  if (tid < 112) {
    float o = Wb2[tid];
    for (int j = 0; j < 256; j++) o = fmaf(hid[j], Ww2[j * 112 + tid], o);
    ow[tid] = o;
  }
  __syncthreads();
  // ---- fcC ----
  {
    float a = Cb1[tid];
#pragma unroll
    for (int k = 0; k < 32; k++) a = fmaf(xcs[k], Cw1[k * 256 + tid], a);
    hid[tid] = tanhf(a);
  }
  __syncthreads();
  if (tid < 64) {
    float o = Cb2[tid];
    for (int j = 0; j < 256; j++) o = fmaf(hid[j], Cw2[j * 64 + tid], o);
    oc[tid] = o;
  }
  __syncthreads();
  // ---- fcR ----
  {
    float a = Rb1[tid];
#pragma unroll
    for (int k = 0; k < 32; k++) a = fmaf(xcs[k], Rw1[k * 256 + tid], a);
    hid[tid] = tanhf(a);
  }
  __syncthreads();
  if (tid < 128) {
    float o = Rb2[tid];
    for (int j = 0; j < 256; j++) o = fmaf(hid[j], Rw2[j * 128 + tid], o);
    orr[tid] = o;
  }
  __syncthreads();

  if (tid < 16) {
    const int h = tid;
    const int p = s * 16 + h;
    float* P = ws + OFF_PAR;
    float kp  = sigmoidf_(ow[h]);
    float ksp = sigmoidf_(ow[16 + h]);
    float kg  = sigmoidf_(ow[32 + h]);
    float gp  = sigmoidf_(ow[48 + h]);
    float gL  = expf(ow[64 + h]) * 2.f;
    float qb  = fmaxf(ow[80 + h], 0.f) + 1e-5f;
    // softmax over nh
    float mx = -3.4e38f;
    for (int j = 0; j < 16; j++) mx = fmaxf(mx, ow[96 + j]);
    float sum = 0.f;
    for (int j = 0; j < 16; j++) sum += expf(ow[96 + j] - mx);
    float ga = expf(ow[96 + h] - mx) / sum;
    float eqs0 = expf(oc[2 * h + 0]) * 10.f;
    float eqs1 = expf(oc[2 * h + 1]) * 10.f;
    float eqg0 = expf(oc[32 + 2 * h + 0]) * 10.f;
    float eqg1 = expf(oc[32 + 2 * h + 1]) * 10.f;
    P[0 * NSNH + p] = kp;   P[1 * NSNH + p] = ksp;
    P[2 * NSNH + p] = kg;   P[3 * NSNH + p] = gp;
    P[4 * NSNH + p] = gL;   P[5 * NSNH + p] = qb;
    P[6 * NSNH + p] = ga;
    P[7 * NSNH + p] = eqs0; P[8 * NSNH + p] = eqs1;
    P[9 * NSNH + p] = eqg0; P[10 * NSNH + p] = eqg1;
    P[11 * NSNH + p] = eqg0 * qb / kg;  // Cg0[0]
    P[12 * NSNH + p] = eqg1 * qb / kg;  // Cg0[1]
    float* RR = ws + OFF_RR;
    for (int j = 0; j < 8; j++) RR[p * 8 + j] = fmaxf(orr[h * 8 + j], 0.f);
  }
}

// =====================================================================
// Kernel 2: big WMMA GEMMs (fcT MLP + fcCT linear) with fused epilogue
//   one wave per 16-row M-tile, 8 waves / block.
//   B matrices pre-swizzled in LDS into K-pair fragment layout so every
//   WMMA operand is a single 8B-aligned, bank-conflict-free ds_load_b64.
// =====================================================================
//   LDS map (floats):
//     W1f : [20 kp][544]   pair-packed fcT_w1 (K 38->40)      ofs 0
//     b1s : [256]                                             ofs 10880
//     W2f : [128 kp][96]   pair-packed fcT_w2                 ofs 11136
//     b2s : [48]                                              ofs 23424
//     Wcf : [18 kp][160]   pair-packed fcCT_w (K 34->36)      ofs 23472
//     bcs : [64]                                              ofs 26352
//     per wave (stride 1792): At [16][42], Hs [16][68], fac[32]  ofs 26416
__global__ __launch_bounds__(256) void k_mlp(
    const float* __restrict__ x, const float* __restrict__ xc,
    const float* __restrict__ Tw1, const float* __restrict__ Tb1,
    const float* __restrict__ Tw2, const float* __restrict__ Tb2,
    const float* __restrict__ Cw, const float* __restrict__ Cb,
    float* __restrict__ ws) {
  __shared__ float sm[40752];
  float* W1f = sm;
  float* b1s = sm + 10880;
  float* W2f = sm + 11136;
  float* b2s = sm + 23424;
  float* Wcf = sm + 23472;
  float* bcs = sm + 26352;
  const int tid = threadIdx.x;

  // ---- stage weights into pair-fragment layouts ----
  for (int i = tid; i < 40 * 256; i += 256) {
    int r = i >> 8, c = i & 255;
    float v = (r < 38) ? Tw1[i] : 0.f;
    W1f[(r >> 1) * 544 + 2 * c + (r & 1)] = v;
  }
  b1s[tid] = Tb1[tid];
  for (int i = tid; i < 256 * 48; i += 256) {
    int r = i / 48, c = i - r * 48;
    W2f[(r >> 1) * 96 + 2 * c + (r & 1)] = Tw2[i];
  }
  if (tid < 48) b2s[tid] = Tb2[tid];
  for (int i = tid; i < 36 * 64; i += 256) {
    int r = i >> 6, c = i & 63;
    float v = (r < 34) ? Cw[i] : 0.f;
    Wcf[(r >> 1) * 160 + 2 * c + (r & 1)] = v;
  }
  if (tid < 64) bcs[tid] = Cb[tid];
  __syncthreads();

  const int wid = tid >> 5;
  const int lane = tid & 31;
  float* At = sm + 26416 + wid * 1792;  // [16][42]
  float* Hs = At + 672;                 // [16][68]
  float* fc = At + 1760;                // [32] per-row factors
  const long long mBase = ((long long)blockIdx.x * 8 + wid) * 16;

  // ---- build A tile: [P,E,T1,T2,R,LAI, xc[0..31], 0, 0] per row ----
  for (int j = 0; j < 20; j++) {
    int idx = j * 32 + lane;
    int r = idx / 40;
    int c = idx - r * 40;
    long long m = mBase + r;
    int sIdx = (int)(m & 1023);
    float v;
    if (c < 6)
      v = x[m * 6 + c];
    else if (c < 38)
      v = xc[sIdx * 32 + (c - 6)];
    else
      v = 0.f;
    At[r * 42 + c] = v;
  }
  // per-row physics factors (snow fraction), lanes 0..15
  if (lane < 16) {
    int r = lane;
    float P = At[r * 42 + 0], E = At[r * 42 + 1];
    float T1 = At[r * 42 + 2], T2 = At[r * 42 + 3];
    float arg = (T1 + T2) / (T2 - T1);
    arg = fminf(fmaxf(arg, -0.999999f), 0.999999f);
    float vf = acosf(arg) / 3.1415f;
    vf = (T1 >= 0.f) ? 0.f : vf;
    vf = (T2 <= 0.f) ? 1.f : vf;
    fc[r] = P * (1.f - vf);  // liquid precip factor
    fc[16 + r] = E;          // evap factor
    ws[OFF_PS + mBase + r] = P * vf;
  }

  const int row = lane & 15;
  const int hi = lane >> 4;  // half-wave id (selects K pair)
  const int koff = hi << 1;

  // layer-2 accumulators (N = 48 -> 3 tiles), init with bias
  v8f vacc[3];
#pragma unroll
  for (int n2 = 0; n2 < 3; n2++) vacc[n2] = splat8(b2s[n2 * 16 + row]);

  for (int chunk = 0; chunk < 4; chunk++) {
    // ---- layer 1: 4 N-tiles of hidden (64 cols), K = 40 (10 steps) ----
#pragma unroll
    for (int nl = 0; nl < 4; nl++) {
      int n = chunk * 4 + nl;
      v8f acc = splat8(b1s[n * 16 + row]);
      for (int ks = 0; ks < 10; ks++) {
        int k0 = ks * 4;
        v2f a = ldv2(At + row * 42 + k0 + koff);
        v2f b = ldv2(W1f + ((k0 >> 1) + hi) * 544 + 2 * (n * 16 + row));
        acc = wmma4(a, b, acc);
      }
#pragma unroll
      for (int i = 0; i < 8; i++) {
        int rr = i + (hi << 3);
        Hs[rr * 68 + nl * 16 + row] = tanhf(acc[i]);
      }
    }
    // ---- layer 2 partial: K-chunk of 64 over 3 N-tiles ----
#pragma unroll
    for (int n2 = 0; n2 < 3; n2++) {
      for (int ks = 0; ks < 16; ks++) {
        int k0 = ks * 4;
        v2f a = ldv2(Hs + row * 68 + k0 + koff);
        v2f b = ldv2(W2f + (chunk * 32 + ks * 2 + hi) * 96 + 2 * (n2 * 16 + row));
        vacc[n2] = wmma4(a, b, vacc[n2]);
      }
    }
  }

  // ---- vc GEMM: A2 = [T1,T2,xc] (K=34->36), N = 64 -> 4 tiles ----
  v8f cacc[4];
#pragma unroll
  for (int nt = 0; nt < 4; nt++) cacc[nt] = splat8(bcs[nt * 16 + row]);
#pragma unroll
  for (int nt = 0; nt < 4; nt++) {
    for (int ks = 0; ks < 9; ks++) {
      int c0 = ks * 4 + koff;          // even A2 K index of this pair
      int mc0 = (c0 < 2) ? (c0 + 2) : (c0 + 4);  // A2 col -> A1 col (pair stays adjacent)
      v2f a = ldv2(At + row * 42 + mc0);
      v2f b = ldv2(Wcf + (ks * 2 + hi) * 160 + 2 * (nt * 16 + row));
      cacc[nt] = wmma4(a, b, cacc[nt]);
    }
  }

  // ---- epilogue v: fuse activations + physics factors ----
  float* PL = ws + OFF_PL;
  float* EV = ws + OFF_EV;
  float* VM = ws + OFF_VM;
#pragma unroll
  for (int n2 = 0; n2 < 3; n2++) {
#pragma unroll
    for (int i = 0; i < 8; i++) {
      int r = i + (hi << 3);
      long long m = mBase + r;
      float v = vacc[n2][i];
      if (n2 == 0) {  // vi = hard_sigmoid(2v); Pl = P(1-vf)*vi
        float vi = fminf(fmaxf((2.f * v + 3.f) * (1.f / 6.f), 0.f), 1.f);
        PL[m * 16 + row] = fc[r] * vi;
      } else if (n2 == 1) {  // ve = relu(v)*2; Ev = E*ve
        EV[m * 16 + row] = fc[16 + r] * (fmaxf(v, 0.f) * 2.f);
      } else {  // vm = exp(v)
        VM[m * 16 + row] = expf(v);
      }
    }
  }
  // ---- epilogue vc: rs/rg = exp(.), interleaved (h,k) layout ----
  float* RS0 = ws + OFF_RS0;
  float* RS1 = ws + OFF_RS1;
  float* RG0 = ws + OFF_RG0;
  float* RG1 = ws + OFF_RG1;
#pragma unroll
  for (int nt = 0; nt < 4; nt++) {
#pragma unroll
    for (int i = 0; i < 8; i++) {
      int r = i + (hi << 3);
      long long m = mBase + r;
      int c = nt * 16 + row;
      float e = expf(cacc[nt][i]);
      if (c < 32) {
        int k = c & 1, hh = c >> 1;
        (k ? RS1 : RS0)[m * 16 + hh] = e;
      } else {
        int cc = c - 32;
        int k = cc & 1, hh = cc >> 1;
        (k ? RG1 : RG0)[m * 16 + hh] = e;
      }
    }
  }
}

// =====================================================================
// Kernel 3: temporal scan + fused causal conv (ring buffers) + h-reduce
//   thread = (site, h); block = 16 sites x 16 h; wave32 shfl reduction
// =====================================================================
__global__ __launch_bounds__(256) void k_scan(const float* __restrict__ ws,
                                              float* __restrict__ out) {
  const int tid = threadIdx.x;
  const int h = tid & 15;
  const int s = blockIdx.x * 16 + (tid >> 4);
  const int p = s * 16 + h;

  const float* P = ws + OFF_PAR;
  const float kp = P[0 * NSNH + p], ksp = P[1 * NSNH + p];
  const float kg = P[2 * NSNH + p], gp = P[3 * NSNH + p];
  const float gL = P[4 * NSNH + p], qb = P[5 * NSNH + p];
  const float ga = P[6 * NSNH + p];
  const float eqs0 = P[7 * NSNH + p], eqs1 = P[8 * NSNH + p];
  const float eqg0 = P[9 * NSNH + p], eqg1 = P[10 * NSNH + p];
  float Cg0v = P[11 * NSNH + p], Cg1v = P[12 * NSNH + p];
  float rw[8];
  {
    const float* RR = ws + OFF_RR;
#pragma unroll
    for (int j = 0; j < 8; j++) rw[j] = RR[p * 8 + j];
  }
  const float* PL = ws + OFF_PL;
  const float* EV = ws + OFF_EV;
  const float* VM = ws + OFF_VM;
  const float* RS0 = ws + OFF_RS0;
  const float* RS1 = ws + OFF_RS1;
  const float* RG0 = ws + OFF_RG0;
  const float* RG1 = ws + OFF_RG1;
  const float* PS = ws + OFF_PS;

  const float qbkg = qb / kg;
  const float omgp = 1.f - gp, kspgp = ksp * gp, ksp1gp = ksp * omgp;
  const float omkg = 1.f - kg;
  float Sf = 0.f, Ssv = 0.f, Sg = 0.f, Cs0 = 0.f, Cs1 = 0.f;
  float qr[8], c0r[8], c1r[8];
#pragma unroll
  for (int j = 0; j < 8; j++) { qr[j] = 0.f; c0r[j] = 0.f; c1r[j] = 0.f; }

  // prefetch t = 0
  float Ps_t = PS[s];
  float Pl_t = PL[p], Ev_t = EV[p], vm_t = VM[p];
  float rs0 = RS0[p], rs1 = RS1[p], rg0 = RG0[p], rg1 = RG1[p];

  for (int t = 0; t < NTt; t++) {
    float nPs = 0.f, nPl = 0.f, nEv = 0.f, nvm = 0.f;
    float nrs0 = 0.f, nrs1 = 0.f, nrg0 = 0.f, nrg1 = 0.f;
    if (t < NTt - 1) {  // software prefetch next step
      long long nb = (long long)(t + 1) * NSNH + p;
      nPs = PS[(long long)(t + 1) * NSs + s];
      nPl = PL[nb]; nEv = EV[nb]; nvm = VM[nb];
      nrs0 = RS0[nb]; nrs1 = RS1[nb]; nrg0 = RG0[nb]; nrg1 = RG1[nb];
    }
    // --- bucket step ---
    float SfP = Sf + Ps_t;
    float qf = fminf(SfP, vm_t);
    Sf = fmaxf(SfP - vm_t, 0.f);
    float H = fmaxf(Ssv + Pl_t + qf - Ev_t, 0.f);
    float qp = fmaxf(kp * (H - gL), 0.f);
    float qs = ksp * fminf(H, gL);
    Ssv = H - qp - qs;
    Cs0 = (Cs0 + rs0 * eqs0 * Ssv) / (1.f + ksp + rs0);
    Cs1 = (Cs1 + rs1 * eqs1 * Ssv) / (1.f + ksp + rs1);
    float qsg = qs * gp;
    float Sgq = Sg + qsg;
    float qg = kg * Sgq + qb;
    Sg = omkg * Sgq - qb;
    float Hg = Sg + qbkg;
    Cg0v = (Cg0v + rg0 * eqg0 * Hg + Cs0 * kspgp) / (1.f + kg + rg0);
    Cg1v = (Cg1v + rg1 * eqg1 * Hg + Cs1 * kspgp) / (1.f + kg + rg1);
    float qsum = qp + qs * omgp + qg;
    float c0 = Cs0 * ksp1gp + Cg0v * kg;
    float c1 = Cs1 * ksp1gp + Cg1v * kg;
    // --- causal conv via ring buffers: y[t] = sum_j x[t-7+j]*w[j] ---
#pragma unroll
    for (int j = 0; j < 7; j++) {
      qr[j] = qr[j + 1]; c0r[j] = c0r[j + 1]; c1r[j] = c1r[j + 1];
    }
    qr[7] = qsum; c0r[7] = c0; c1r[7] = c1;
    float QR = 0.f, C0 = 0.f, C1 = 0.f;
#pragma unroll
    for (int j = 0; j < 8; j++) {
      QR = fmaf(qr[j], rw[j], QR);
      C0 = fmaf(c0r[j], rw[j], C0);
      C1 = fmaf(c1r[j], rw[j], C1);
    }
    // --- reduce over h (16 lanes of the wave32 half) ---
    float aQ = QR * ga, a0 = C0 * ga, a1 = C1 * ga;
#pragma unroll
    for (int msk = 1; msk < 16; msk <<= 1) {
      aQ += __shfl_xor(aQ, msk, 16);
      a0 += __shfl_xor(a0, msk, 16);
      a1 += __shfl_xor(a1, msk, 16);
    }
    if (h == 0) {
      long long o = (long long)t * NSs + s;
      out[o] = aQ;
      out[MTOT + o * 2 + 0] = a0 / aQ;
      out[MTOT + o * 2 + 1] = a1 / aQ;
    }
    Ps_t = nPs; Pl_t = nPl; Ev_t = nEv; vm_t = nvm;
    rs0 = nrs0; rs1 = nrs1; rg0 = nrg0; rg1 = nrg1;
  }
}

// =====================================================================
extern "C" void kernel_launch(void* const* d_in, const int* in_sizes, int n_in,
                              void* d_out, int out_size, void* d_ws,
                              size_t ws_size, hipStream_t stream) {
  (void)in_sizes; (void)n_in; (void)out_size; (void)ws_size;
  const float* x    = (const float*)d_in[0];
  const float* xc   = (const float*)d_in[1];
  const float* Rw1  = (const float*)d_in[2];
  const float* Rb1  = (const float*)d_in[3];
  const float* Rw2  = (const float*)d_in[4];
  const float* Rb2  = (const float*)d_in[5];
  const float* Ww1  = (const float*)d_in[6];
  const float* Wb1  = (const float*)d_in[7];
  const float* Ww2  = (const float*)d_in[8];
  const float* Wb2  = (const float*)d_in[9];
  const float* Tw1  = (const float*)d_in[10];
  const float* Tb1  = (const float*)d_in[11];
  const float* Tw2  = (const float*)d_in[12];
  const float* Tb2  = (const float*)d_in[13];
  const float* Cw1  = (const float*)d_in[14];
  const float* Cb1  = (const float*)d_in[15];
  const float* Cw2  = (const float*)d_in[16];
  const float* Cb2  = (const float*)d_in[17];
  const float* CTw  = (const float*)d_in[18];
  const float* CTb  = (const float*)d_in[19];
  float* ws = (float*)d_ws;
  float* out = (float*)d_out;

  k_site<<<NSs, 256, 0, stream>>>(xc, Rw1, Rb1, Rw2, Rb2, Ww1, Wb1, Ww2, Wb2,
                                  Cw1, Cb1, Cw2, Cb2, ws);
  k_mlp<<<(int)(MTOT / 128), 256, 0, stream>>>(x, xc, Tw1, Tb1, Tw2, Tb2, CTw,
                                               CTb, ws);
  k_scan<<<NSNH / 256, 256, 0, stream>>>(ws, out);
}